// Kernel_4647922305226660848_53472342835839
// MI455X (gfx1250) — compile-verified
//
#include <hip/hip_runtime.h>
#include <hip/hip_bf16.h>

// MI455X / gfx1250, wave32. fp32 pipeline on V_WMMA_F32_16X16X4_F32.
// LDS operands stored [row][k] with k-stride padded to 34 so every A/B
// fragment is one aligned ds_load_b64 and fragment reads spread across banks.

#define NS   32
#define CC   128
#define HH   56
#define WW   56
#define HWP  3136        // 56*56
#define C3   384         // 3*C
#define PADK 34          // padded k-stride (even -> float2 aligned)

typedef float v2f __attribute__((ext_vector_type(2)));
typedef float v8f __attribute__((ext_vector_type(8)));

__device__ __forceinline__ v8f wmma_f32(v2f a, v2f b, v8f c) {
    // (neg_a, A, neg_b, B, c_mod, C, reuse_a, reuse_b)
    return __builtin_amdgcn_wmma_f32_16x16x4_f32(false, a, false, b, (short)0, c, false, false);
}

// ---------------------------------------------------------------------------
// GEMM1:  t3[c,pos] = sum_j p3[c,j] * T1[j,pos]
//         T1[j=3*ci+tap, pos] = x[ci, h+2*(tap-1), w]  (vertical, dil=2)
// Block tile 128(M) x 224(N), K steps of 32. Wave grid 4(M-rows) x 2(N-cols);
// each wave: 2 M-subtiles x 7 N-subtiles -> 14 WMMA per k-substep.
// ---------------------------------------------------------------------------
__global__ __launch_bounds__(256) void k_gemm1(const float* __restrict__ p3,
                                               const float* __restrict__ x,
                                               float* __restrict__ t3) {
    __shared__ float As[128 * PADK];   // [m][k]
    __shared__ float Bs[224 * PADK];   // [n][k]
    const int t    = threadIdx.x;
    const int lane = t & 31;
    const int wv   = t >> 5;
    const int smp  = blockIdx.y;
    const int pos0 = blockIdx.x * 224;
    const float* xs = x + (size_t)smp * CC * HWP;

    const int mn    = lane & 15;
    const int kp    = (lane >> 4) << 1;   // 0 or 2
    const int rmsb  = (lane >> 4) << 3;   // 0 or 8
    const int mbase = (wv >> 1) * 32;     // 4 wave-rows of 32
    const int nbase = (wv & 1) * 112;     // 2 wave-cols of 112

    v8f acc[2][7] = {};

    for (int ks = 0; ks < C3; ks += 32) {
        #pragma unroll
        for (int i = 0; i < 16; ++i) {               // A: 128x32 dense
            int idx = t + i * 256;
            int m = idx >> 5, k = idx & 31;
            As[m * PADK + k] = p3[m * C3 + ks + k];
        }
        #pragma unroll
        for (int i = 0; i < 28; ++i) {               // B: 32x224 gathered from x
            int idx = t + i * 256;                   // coalesced along nn
            int k = idx / 224, nn = idx % 224;
            int j = ks + k;
            int tap = j % 3, ci = j / 3;
            int pos = pos0 + nn;
            int hp = pos / WW + 2 * (tap - 1);
            float v = 0.0f;
            if ((unsigned)hp < (unsigned)HH)
                v = xs[ci * HWP + pos + 2 * WW * (tap - 1)];
            Bs[nn * PADK + k] = v;
        }
        __syncthreads();
        #pragma unroll
        for (int kk = 0; kk < 8; ++kk) {
            int ko = kk * 4 + kp;
            v2f a0 = *(const v2f*)&As[(mbase + mn) * PADK + ko];
            v2f a1 = *(const v2f*)&As[(mbase + 16 + mn) * PADK + ko];
            #pragma unroll
            for (int ni = 0; ni < 7; ++ni) {
                v2f b = *(const v2f*)&Bs[(nbase + ni * 16 + mn) * PADK + ko];
                acc[0][ni] = wmma_f32(a0, b, acc[0][ni]);
                acc[1][ni] = wmma_f32(a1, b, acc[1][ni]);
            }
        }
        __syncthreads();
    }

    float* o = t3 + (size_t)smp * CC * HWP;
    #pragma unroll
    for (int mi = 0; mi < 2; ++mi) {
        #pragma unroll
        for (int ni = 0; ni < 7; ++ni) {
            int col = pos0 + nbase + ni * 16 + mn;
            #pragma unroll
            for (int r = 0; r < 8; ++r) {
                int m = mbase + mi * 16 + r + rmsb;
                o[m * HWP + col] = acc[mi][ni][r];
            }
        }
    }
}

// ---------------------------------------------------------------------------
// Elementwise: t12[j,pos] = p5[pos]*t3[j%C,pos] - p2[j%3,w]*T1[j,pos]
//                           - p10[j,pos]*softmax3(x horiz, dil=2)
// ---------------------------------------------------------------------------
__global__ __launch_bounds__(256) void k_t12(const float* __restrict__ x,
                                             const float* __restrict__ t3,
                                             const float* __restrict__ p2,
                                             const float* __restrict__ p5,
                                             const float* __restrict__ p10,
                                             float* __restrict__ t12) {
    int tid  = blockIdx.x * 256 + threadIdx.x;
    int pos  = tid % HWP;
    int rest = tid / HWP;
    int c    = rest % CC;
    int smp  = rest / CC;
    int h = pos / WW, w = pos % WW;

    const float* xc = x + ((size_t)smp * CC + c) * HWP;
    float x0  = xc[pos];
    float xvm = (h >= 2)      ? xc[pos - 2 * WW] : 0.0f;
    float xvp = (h < HH - 2)  ? xc[pos + 2 * WW] : 0.0f;
    float xhm = (w >= 2)      ? xc[pos - 2]      : 0.0f;
    float xhp = (w < WW - 2)  ? xc[pos + 2]      : 0.0f;

    float mx = fmaxf(xhm, fmaxf(x0, xhp));
    float e0 = __expf(xhm - mx), e1 = __expf(x0 - mx), e2 = __expf(xhp - mx);
    float inv = 1.0f / (e0 + e1 + e2);
    float t8[3] = { e0 * inv, e1 * inv, e2 * inv };
    float xv[3] = { xvm, x0, xvp };

    float p5v = p5[pos];
    const float* t3s  = t3  + (size_t)smp * CC * HWP;
    const float* p10s = p10 + (size_t)(3 * c) * HWP + pos;
    float* outj = t12 + ((size_t)smp * C3 + 3 * c) * HWP + pos;

    #pragma unroll
    for (int k = 0; k < 3; ++k) {
        int j = 3 * c + k;
        int b = j & (CC - 1);                      // j % 128
        float t5v  = p5v * t3s[b * HWP + pos];
        float t2v  = p2[k * WW + w] * xv[k];
        float t10v = p10s[k * HWP] * t8[k];
        outj[k * HWP] = t5v - t2v - t10v;
    }
}

// ---------------------------------------------------------------------------
// GEMM2:  t9[c,d] = (1/56) * sum_pos x[c,pos] * T4[d,pos]
//         T4[d=3*ci+tap, pos] = t3[ci, h, w+(tap-1)]   (horizontal, dil=1)
// Block tile 128(M) x 128(N), K=3136 in steps of 32. Wave grid 4x2;
// each wave: 2 M-subtiles x 4 N-subtiles.
// ---------------------------------------------------------------------------
__global__ __launch_bounds__(256) void k_gemm2(const float* __restrict__ x,
                                               const float* __restrict__ t3,
                                               float* __restrict__ t9) {
    __shared__ float As[128 * PADK];   // [m][k]
    __shared__ float Bs[128 * PADK];   // [n][k]
    const int t    = threadIdx.x;
    const int lane = t & 31;
    const int wv   = t >> 5;
    const int smp  = blockIdx.y;
    const int d0   = blockIdx.x * 128;
    const float* xs  = x  + (size_t)smp * CC * HWP;
    const float* t3s = t3 + (size_t)smp * CC * HWP;

    const int mn    = lane & 15;
    const int kp    = (lane >> 4) << 1;
    const int rmsb  = (lane >> 4) << 3;
    const int mbase = (wv >> 1) * 32;
    const int nbase = (wv & 1) * 64;

    v8f acc[2][4] = {};

    for (int ks = 0; ks < HWP; ks += 32) {
        #pragma unroll
        for (int i = 0; i < 16; ++i) {               // A: x rows, 128x32
            int idx = t + i * 256;
            int m = idx >> 5, k = idx & 31;
            As[m * PADK + k] = xs[m * HWP + ks + k];
        }
        #pragma unroll
        for (int i = 0; i < 16; ++i) {               // B: 128x32 gathered from t3
            int idx = t + i * 256;                   // coalesced along k (=pos)
            int n = idx >> 5, k = idx & 31;
            int d = d0 + n;
            int tap = d % 3, ci = d / 3;
            int pos = ks + k;
            int wp = pos % WW + (tap - 1);
            float v = 0.0f;
            if ((unsigned)wp < (unsigned)WW)
                v = t3s[ci * HWP + pos + (tap - 1)];
            Bs[n * PADK + k] = v;
        }
        __syncthreads();
        #pragma unroll
        for (int kk = 0; kk < 8; ++kk) {
            int ko = kk * 4 + kp;
            v2f a0 = *(const v2f*)&As[(mbase + mn) * PADK + ko];
            v2f a1 = *(const v2f*)&As[(mbase + 16 + mn) * PADK + ko];
            #pragma unroll
            for (int ni = 0; ni < 4; ++ni) {
                v2f b = *(const v2f*)&Bs[(nbase + ni * 16 + mn) * PADK + ko];
                acc[0][ni] = wmma_f32(a0, b, acc[0][ni]);
                acc[1][ni] = wmma_f32(a1, b, acc[1][ni]);
            }
        }
        __syncthreads();
    }

    const float scale = 1.0f / 56.0f;               // 1/sqrt(HW)
    float* out = t9 + (size_t)smp * CC * C3;
    #pragma unroll
    for (int mi = 0; mi < 2; ++mi) {
        #pragma unroll
        for (int ni = 0; ni < 4; ++ni) {
            int col = d0 + nbase + ni * 16 + mn;
            #pragma unroll
            for (int r = 0; r < 8; ++r) {
                int m = mbase + mi * 16 + r + rmsb;
                out[m * C3 + col] = acc[mi][ni][r] * scale;
            }
        }
    }
}

// ---------------------------------------------------------------------------
// GEMM3:  out[c,pos] = (1/sqrt(384)) * sum_d t9[c,d] * t12[d,pos]
// Block tile 128(M) x 224(N), K=384 in steps of 32. Dense operands.
// ---------------------------------------------------------------------------
__global__ __launch_bounds__(256) void k_gemm3(const float* __restrict__ t9,
                                               const float* __restrict__ t12,
                                               float* __restrict__ out) {
    __shared__ float As[128 * PADK];   // [m][k]
    __shared__ float Bs[224 * PADK];   // [n][k]
    const int t    = threadIdx.x;
    const int lane = t & 31;
    const int wv   = t >> 5;
    const int smp  = blockIdx.y;
    const int pos0 = blockIdx.x * 224;
    const float* t9s  = t9  + (size_t)smp * CC * C3;
    const float* t12s = t12 + (size_t)smp * C3 * HWP;

    const int mn    = lane & 15;
    const int kp    = (lane >> 4) << 1;
    const int rmsb  = (lane >> 4) << 3;
    const int mbase = (wv >> 1) * 32;
    const int nbase = (wv & 1) * 112;

    v8f acc[2][7] = {};

    for (int ks = 0; ks < C3; ks += 32) {
        #pragma unroll
        for (int i = 0; i < 16; ++i) {               // A: t9, 128x32
            int idx = t + i * 256;
            int m = idx >> 5, k = idx & 31;
            As[m * PADK + k] = t9s[m * C3 + ks + k];
        }
        #pragma unroll
        for (int i = 0; i < 28; ++i) {               // B: t12, 32x224 dense
            int idx = t + i * 256;                   // coalesced along nn
            int k = idx / 224, nn = idx % 224;
            Bs[nn * PADK + k] = t12s[(ks + k) * HWP + pos0 + nn];
        }
        __syncthreads();
        #pragma unroll
        for (int kk = 0; kk < 8; ++kk) {
            int ko = kk * 4 + kp;
            v2f a0 = *(const v2f*)&As[(mbase + mn) * PADK + ko];
            v2f a1 = *(const v2f*)&As[(mbase + 16 + mn) * PADK + ko];
            #pragma unroll
            for (int ni = 0; ni < 7; ++ni) {
                v2f b = *(const v2f*)&Bs[(nbase + ni * 16 + mn) * PADK + ko];
                acc[0][ni] = wmma_f32(a0, b, acc[0][ni]);
                acc[1][ni] = wmma_f32(a1, b, acc[1][ni]);
            }
        }
        __syncthreads();
    }

    const float scale = 0.051031036307982884f;      // 1/sqrt(384)
    float* o = out + (size_t)smp * CC * HWP;
    #pragma unroll
    for (int mi = 0; mi < 2; ++mi) {
        #pragma unroll
        for (int ni = 0; ni < 7; ++ni) {
            int col = pos0 + nbase + ni * 16 + mn;
            #pragma unroll
            for (int r = 0; r < 8; ++r) {
                int m = mbase + mi * 16 + r + rmsb;
                o[m * HWP + col] = acc[mi][ni][r] * scale;
            }
        }
    }
}

// ---------------------------------------------------------------------------
extern "C" void kernel_launch(void* const* d_in, const int* in_sizes, int n_in,
                              void* d_out, int out_size, void* d_ws, size_t ws_size,
                              hipStream_t stream) {
    (void)in_sizes; (void)n_in; (void)out_size; (void)ws_size;
    const float* x   = (const float*)d_in[0];   // (32,128,56,56)
    const float* p2  = (const float*)d_in[1];   // (1,1,3,1,56)
    const float* p3  = (const float*)d_in[2];   // (128,384)
    const float* p5  = (const float*)d_in[3];   // (1,1,56,56)
    const float* p10 = (const float*)d_in[4];   // (1,128,3,56,56)
    float* out = (float*)d_out;

    char* ws = (char*)d_ws;
    const size_t t3_bytes  = (size_t)NS * CC * HWP * sizeof(float);   // 51.4 MB
    const size_t t12_bytes = (size_t)NS * C3 * HWP * sizeof(float);   // 154.1 MB
    float* t3  = (float*)ws;
    float* t12 = (float*)(ws + t3_bytes);
    float* t9  = (float*)(ws + t3_bytes + t12_bytes);                 // 6.3 MB

    dim3 blk(256);
    k_gemm1<<<dim3(HWP / 224, NS), blk, 0, stream>>>(p3, x, t3);
    k_t12 <<<dim3((NS * CC * HWP) / 256), blk, 0, stream>>>(x, t3, p2, p5, p10, t12);
    k_gemm2<<<dim3(C3 / 128, NS), blk, 0, stream>>>(x, t3, t9);
    k_gemm3<<<dim3(HWP / 224, NS), blk, 0, stream>>>(t9, t12, out);
}